// patch_loss_32177894982199
// MI455X (gfx1250) — compile-verified
//
#include <hip/hip_runtime.h>

// MI455X / gfx1250, wave32. One wave per (b,c,n) patch unit.
// Per-patch sum / sum-of-squares reductions are done on the matrix pipe with
// V_WMMA_F32_16X16X4_F32 (A = ones 16x4, chained 4x over the 16 patch rows),
// leaving only a 4-step shfl_xor to collapse the 16 column sums.

typedef float v2f __attribute__((ext_vector_type(2)));
typedef float v8f __attribute__((ext_vector_type(8)));

#define PATCHK 16
#define BB 4
#define CC 3
#define HH 512
#define WW 512
#define NN 4096
#define PP 256                      // pixels per patch
#define UNITS (BB * CC * NN)        // 49152 waves
#define INV_TOTAL (1.0f / (float)(BB * CC * NN * PP))

__global__ __launch_bounds__(256) void patch_loss_kernel(
    const float* __restrict__ fuse,
    const float* __restrict__ img1,
    const float* __restrict__ img2,
    const int*   __restrict__ coords,
    float*       __restrict__ out)
{
    const int lane = threadIdx.x & 31;
    const int unit = blockIdx.x * 8 + (threadIdx.x >> 5);   // (b,c,n)

    const int b   = unit / (CC * NN);
    const int rem = unit - b * (CC * NN);
    const int c   = rem / NN;
    const int n   = rem - c * NN;

    // uniform per-wave coords (broadcast load)
    const int h = coords[((size_t)b * NN + n) * 2 + 0];
    const int w = coords[((size_t)b * NN + n) * 2 + 1];

    // lane -> pixel mapping: col = lane&15, rows 4s + 2v + half (s=0..3, v=0..1)
    const int half = lane >> 4;
    const int col  = lane & 15;
    const size_t base = ((size_t)(b * CC + c) * HH + (size_t)h) * WW + (size_t)w + col;

    float x1[4][2], x2[4][2], xf[4][2];
#pragma unroll
    for (int s = 0; s < 4; ++s) {
#pragma unroll
        for (int v = 0; v < 2; ++v) {
            const size_t a = base + (size_t)(4 * s + 2 * v + half) * WW;
            x1[s][v] = img1[a];
            x2[s][v] = img2[a];
            xf[s][v] = fuse[a];
        }
    }

    // ---- WMMA reduction: D = ones(16x4) x B(4x16) + C, chained over 4 steps.
    // With A == 1, D accumulates the sum of every element fed through B;
    // after 4 steps D holds the 16 per-column sums (replicated across M).
    v2f ones; ones[0] = 1.0f; ones[1] = 1.0f;
    v8f a1 = {}; v8f aq1 = {}; v8f a2 = {}; v8f aq2 = {};
#pragma unroll
    for (int s = 0; s < 4; ++s) {
        v2f b1;  b1[0]  = x1[s][0];             b1[1]  = x1[s][1];
        v2f bq1; bq1[0] = x1[s][0] * x1[s][0];  bq1[1] = x1[s][1] * x1[s][1];
        v2f b2;  b2[0]  = x2[s][0];             b2[1]  = x2[s][1];
        v2f bq2; bq2[0] = x2[s][0] * x2[s][0];  bq2[1] = x2[s][1] * x2[s][1];
        a1  = __builtin_amdgcn_wmma_f32_16x16x4_f32(false, ones, false, b1,
                                                    (short)0, a1,  false, false);
        aq1 = __builtin_amdgcn_wmma_f32_16x16x4_f32(false, ones, false, bq1,
                                                    (short)0, aq1, false, false);
        a2  = __builtin_amdgcn_wmma_f32_16x16x4_f32(false, ones, false, b2,
                                                    (short)0, a2,  false, false);
        aq2 = __builtin_amdgcn_wmma_f32_16x16x4_f32(false, ones, false, bq2,
                                                    (short)0, aq2, false, false);
    }

    // Each lane's VGPR0 of D holds one column sum; each 16-lane half holds all
    // 16 columns, so a 4-step xor reduce inside the half gives the patch total.
    float t1  = a1[0];
    float tq1 = aq1[0];
    float t2  = a2[0];
    float tq2 = aq2[0];
#pragma unroll
    for (int off = 1; off < 16; off <<= 1) {
        t1  += __shfl_xor(t1,  off, 32);
        tq1 += __shfl_xor(tq1, off, 32);
        t2  += __shfl_xor(t2,  off, 32);
        tq2 += __shfl_xor(tq2, off, 32);
    }

    const float invP = 1.0f / (float)PP;
    const float m1 = t1 * invP;
    const float m2 = t2 * invP;
    const float sd1 = sqrtf(fmaxf(tq1 * invP - m1 * m1, 0.0f));
    const float sd2 = sqrtf(fmaxf(tq2 * invP - m2 * m2, 0.0f));
    const float dn  = sd1 + sd2 + 1e-6f;
    const float w1  = sd1 / dn;
    const float w2  = sd2 / dn;

    // second pass over register-resident pixels
    float acc = 0.0f;
#pragma unroll
    for (int s = 0; s < 4; ++s) {
#pragma unroll
        for (int v = 0; v < 2; ++v) {
            const float tgt = fmaf(w1, x1[s][v], w2 * x2[s][v]);
            acc += fabsf(xf[s][v] - tgt);
        }
    }
#pragma unroll
    for (int off = 1; off < 32; off <<= 1)
        acc += __shfl_xor(acc, off, 32);

    if (lane == 0)
        atomicAdd(out, acc * INV_TOTAL);
}

extern "C" void kernel_launch(void* const* d_in, const int* in_sizes, int n_in,
                              void* d_out, int out_size, void* d_ws, size_t ws_size,
                              hipStream_t stream) {
    const float* fuse   = (const float*)d_in[0];
    const float* img1   = (const float*)d_in[1];
    const float* img2   = (const float*)d_in[2];
    const int*   coords = (const int*)d_in[3];
    float*       out    = (float*)d_out;

    hipMemsetAsync(out, 0, sizeof(float), stream);   // zero the scalar accumulator

    const int blocks = UNITS / 8;                    // 8 waves (256 threads) per block
    patch_loss_kernel<<<blocks, 256, 0, stream>>>(fuse, img1, img2, coords, out);
}